// EMLTreeVectorized_67946382623157
// MI455X (gfx1250) — compile-verified
//
#include <hip/hip_runtime.h>
#include <hip/hip_bf16.h>
#include <stdint.h>

// ============================================================================
// EML soft decision tree forward, MI455X (gfx1250).
//
// Pure VALU/TRANS workload (~2.6 G f32 ops, ~1.4 MB total traffic @ 23.3TB/s
// HBM => compute-latency bound; no matrix structure => WMMA inapplicable).
// CDNA5 paths used: GLOBAL_LOAD_ASYNC_TO_LDS_B128 (ASYNCcnt DMA, via inline
// asm since the toolchain builtin signature is nonstandard) to stage the
// per-tree logits tile into LDS, s_wait_asynccnt, wave32 block of 8 waves,
// v_exp/v_log/v_rcp transcendentals, split barrier via __syncthreads.
// ============================================================================

#define BLOCK 256
#define BDIM  2048
#define TDIM  128

#if defined(__has_builtin)
#  if __has_builtin(__builtin_amdgcn_s_wait_asynccnt)
#    define HAVE_WAIT_ASYNC 1
#  endif
#endif

typedef __attribute__((address_space(3))) void LVoid;

// CDNA5 async global->LDS DMA (tracked by ASYNCcnt). Inline asm: the clang
// builtin's signature differs between toolchains, asm is portable.
__device__ __forceinline__ void async_g2l_b128(const float* g, float* l) {
  unsigned int       laddr = (unsigned int)(unsigned long long)(LVoid*)l;
  unsigned long long gaddr = (unsigned long long)g;
  asm volatile("global_load_async_to_lds_b128 %0, %1, off"
               :: "v"(laddr), "v"(gaddr) : "memory");
}

__device__ __forceinline__ void wait_async_zero() {
#ifdef HAVE_WAIT_ASYNC
  __builtin_amdgcn_s_wait_asynccnt(0);
#else
  asm volatile("s_wait_asynccnt 0" ::: "memory");
#endif
}

// ---- deterministic counter-based uniform in (0,1) (PCG output hash) -------
__device__ __forceinline__ float unit_uniform(uint32_t v) {
  uint32_t s = v * 747796405u + 2891336453u;
  uint32_t w = ((s >> ((s >> 28u) + 4u)) ^ s) * 277803737u;
  w = (w >> 22u) ^ w;
  return ((float)(w >> 8) + 0.5f) * (1.0f / 16777216.0f);  // (2^-25, 1)
}

static __device__ constexpr float INVK[17] = {
  0.0f, 1.0f, 1.0f/2.0f, 1.0f/3.0f, 1.0f/4.0f, 1.0f/5.0f, 1.0f/6.0f,
  1.0f/7.0f, 1.0f/8.0f, 1.0f/9.0f, 1.0f/10.0f, 1.0f/11.0f, 1.0f/12.0f,
  1.0f/13.0f, 1.0f/14.0f, 1.0f/15.0f, 1.0f/16.0f };

// exp Taylor (order 12) + ln Taylor (order 16), reference semantics
__device__ __forceinline__ float eml_fn(float ex, float ey) {
  ex = fminf(5.0f, fmaxf(-5.0f, ex));
  ey = fmaxf(1e-30f, ey);
  float r = 1.0f;
#pragma unroll
  for (int k = 12; k >= 1; --k) r = fmaf(ex * r, INVK[k], 1.0f);   // 1 + x*r/k
  float z = ey - 1.0f;
  float q = 0.0f;
#pragma unroll
  for (int k = 16; k >= 1; --k) q = fmaf(-z, q, INVK[k]);          // 1/k - z*q
  return r + z * q;
}

// One softmax-weighted candidate sum. With tau==1:
//   w_c ∝ exp(l_c + g_c) = exp(l_c) / E_c,  E_c = -ln(u_c) ~ Exp(1)
// s_exp holds exp(logits) (precomputed once per tree t, shared by all b).
template <int NCH>
__device__ __forceinline__ float eval_side(const float* __restrict__ s_exp,
                                           int base, uint32_t hib, uint32_t bt,
                                           const float4& xv, float prevc) {
  float Z = 0.0f, acc = 0.0f;
#pragma unroll
  for (int c = 0; c < NCH; ++c) {
    float e = s_exp[base + 2 * c];
    float u = unit_uniform(((hib + (uint32_t)c) << 18) + bt);
    float E = __builtin_amdgcn_logf(u) * (-0.6931471805599453f);  // -ln(u) > 0
    float s = e * __builtin_amdgcn_rcpf(E);
    Z += s;
    float cand = (c == 0) ? 1.0f :
                 (c == 1) ? xv.x : (c == 2) ? xv.y :
                 (c == 3) ? xv.z : (c == 4) ? xv.w : prevc;
    acc = fmaf(cand, s, acc);
  }
  return acc * __builtin_amdgcn_rcpf(Z);
}

__device__ __forceinline__ void stage_level(const float* g, float* lds,
                                            int count_f32, int tid) {
  const int n16 = count_f32 >> 2;            // all levels are multiples of 4
  for (int i = tid; i < n16; i += BLOCK)
    async_g2l_b128(g + 4 * i, lds + 4 * i);
}

__device__ __forceinline__ void level_pass(const float* __restrict__ s_exp,
                                           float* __restrict__ s_prev, int tid,
                                           uint32_t bt, const float4& xv,
                                           int nn, int loff, int nbase) {
  for (int n = 0; n < nn; ++n) {
    float pl = s_prev[(2 * n)     * BLOCK + tid];
    float pr = s_prev[(2 * n + 1) * BLOCK + tid];
    int      base = loff + n * 12;                       // 6 choices * 2 sides
    uint32_t hib  = (uint32_t)(((nbase + n) * 2) * 8);   // side 0
    float ex = eval_side<6>(s_exp, base + 0, hib,     bt, xv, pl);
    float ey = eval_side<6>(s_exp, base + 1, hib + 8, bt, xv, pr);
    s_prev[n * BLOCK + tid] = eml_fn(ex, ey);
  }
}

__global__ __launch_bounds__(BLOCK)
void eml_tree_fwd_kernel(const float* __restrict__ x,
                         const float* __restrict__ l0, const float* __restrict__ l1,
                         const float* __restrict__ l2, const float* __restrict__ l3,
                         const float* __restrict__ l4, const float* __restrict__ l5,
                         float* __restrict__ out) {
  // per-tree logits tile (floats): l0:32*5*2=320, l1:192, l2:96, l3:48, l4:24, l5:12
  __shared__ __align__(16) float s_exp[692];
  __shared__ float s_prev[32 * BLOCK];     // per-thread node frontier, 32 KB

  const int tid = threadIdx.x;
  const int t   = blockIdx.y;
  const int b   = blockIdx.x * BLOCK + tid;

  // ---- stage this tree's logits into LDS via CDNA5 async DMA -------------
  stage_level(l0 + t * 320,  s_exp + 0,   320, tid);
  stage_level(l1 + t * 192,  s_exp + 320, 192, tid);
  stage_level(l2 + t * 96,   s_exp + 512,  96, tid);
  stage_level(l3 + t * 48,   s_exp + 608,  48, tid);
  stage_level(l4 + t * 24,   s_exp + 656,  24, tid);
  stage_level(l5 + t * 12,   s_exp + 680,  12, tid);
  wait_async_zero();          // this wave's async copies landed in LDS
  __syncthreads();            // all waves' copies visible

  // ---- softmax numerators exp(l) once per tree (shared by 2048 b rows) ---
  for (int i = tid; i < 692; i += BLOCK)
    s_exp[i] = __builtin_amdgcn_exp2f(s_exp[i] * 1.4426950408889634f);
  __syncthreads();

  const float4   xv = reinterpret_cast<const float4*>(x)[b];
  const uint32_t bt = (uint32_t)(t * BDIM + b);

  // level 0: 32 nodes, 5 choices, no child candidate
  for (int n = 0; n < 32; ++n) {
    int      base = n * 10;
    uint32_t hib  = (uint32_t)((n * 2) * 8);
    float ex = eval_side<5>(s_exp, base + 0, hib,     bt, xv, 0.0f);
    float ey = eval_side<5>(s_exp, base + 1, hib + 8, bt, xv, 0.0f);
    s_prev[n * BLOCK + tid] = eml_fn(ex, ey);
  }
  // levels 1..4 (in-place frontier halving; per-thread column => no races)
  level_pass(s_exp, s_prev, tid, bt, xv, 16, 320, 32);
  level_pass(s_exp, s_prev, tid, bt, xv,  8, 512, 48);
  level_pass(s_exp, s_prev, tid, bt, xv,  4, 608, 56);
  level_pass(s_exp, s_prev, tid, bt, xv,  2, 656, 60);
  // level 5: single root node -> output
  {
    float pl = s_prev[0 * BLOCK + tid];
    float pr = s_prev[1 * BLOCK + tid];
    uint32_t hib = (uint32_t)((62 * 2) * 8);
    float ex = eval_side<6>(s_exp, 680, hib,     bt, xv, pl);
    float ey = eval_side<6>(s_exp, 681, hib + 8, bt, xv, pr);
    out[(size_t)b * TDIM + t] = eml_fn(ex, ey);
  }
}

extern "C" void kernel_launch(void* const* d_in, const int* in_sizes, int n_in,
                              void* d_out, int out_size, void* d_ws, size_t ws_size,
                              hipStream_t stream) {
  const float* x  = (const float*)d_in[0];
  const float* l0 = (const float*)d_in[1];
  const float* l1 = (const float*)d_in[2];
  const float* l2 = (const float*)d_in[3];
  const float* l3 = (const float*)d_in[4];
  const float* l4 = (const float*)d_in[5];
  const float* l5 = (const float*)d_in[6];
  float* out = (float*)d_out;

  dim3 grid(BDIM / BLOCK, TDIM);   // (8, 128) blocks of 256 threads (8 waves)
  eml_tree_fwd_kernel<<<grid, BLOCK, 0, stream>>>(x, l0, l1, l2, l3, l4, l5, out);
}